// CustomConv2D_12953621365171
// MI455X (gfx1250) — compile-verified
//
#include <hip/hip_runtime.h>

typedef __attribute__((ext_vector_type(2))) float v2f;
typedef __attribute__((ext_vector_type(8))) float v8f;

#define CIN   64
#define COUT  64
#define HOUT  112
#define HPAD  114   // 112 + 2*PAD

// One wave computes a 16(co) x 112(w) strip of one output row h of one image b,
// as 7 tiles of 16x16 via V_WMMA_F32_16X16X4_F32 (K = 9 padded to 12 -> 3 WMMAs).
__global__ __launch_bounds__(256) void conv3x3_lastchan_wmma(
    const float* __restrict__ x,     // (B, CIN, 114, 114)
    const float* __restrict__ wgt,   // (COUT, CIN, 3, 3)
    const float* __restrict__ bias,  // (COUT)
    float* __restrict__ out)         // (B, COUT, 112, 112)
{
    const int lane  = threadIdx.x & 31;
    const int wib   = threadIdx.x >> 5;                 // wave in block (0..7)
    const int gw    = blockIdx.x * 8 + wib;             // global wave id

    // gw -> (b, co_tile, h):  waves = 64 * 4 * 112
    const int h   = gw % HOUT;
    const int cot = (gw / HOUT) & 3;
    const int b   = gw / (HOUT * 4);

    const int m     = lane & 15;                        // M (A) / N (B,C) index
    const int halfk = (lane >= 16) ? 2 : 0;             // K sub-lane split

    // ---- A fragment: weights of channel CIN-1, 16 co x 12 K (9 real + 3 zero pad)
    // weight[co][63][kh][kw] -> flat co*576 + 567 + k, k = kh*3+kw
    const float* wrow = wgt + (size_t)(cot * 16 + m) * (CIN * 9) + (size_t)(CIN - 1) * 9;
    v2f a[3];
#pragma unroll
    for (int c = 0; c < 3; ++c) {
        const int k0 = c * 4 + halfk;
        const int kx = (k0     < 9) ? k0     : 8;       // clamped, always in-bounds
        const int ky = (k0 + 1 < 9) ? k0 + 1 : 8;
        float ax = wrow[kx];
        float ay = wrow[ky];
        a[c].x = (k0     < 9) ? ax : 0.0f;              // zero the K padding
        a[c].y = (k0 + 1 < 9) ? ay : 0.0f;
    }

    // ---- Bias-seeded accumulator template (C layout: VGPR r -> co = base + r (+8))
    const int coHalf = cot * 16 + ((lane >= 16) ? 8 : 0);
    v8f cbias;
#pragma unroll
    for (int r = 0; r < 8; ++r) cbias[r] = bias[coHalf + r];

    // ---- Input plane: channel CIN-1 of image b (114x114, L2-resident: 3.3 MB total)
    const float* xplane = x + (size_t)(b * CIN + (CIN - 1)) * HPAD * HPAD;

    // ---- Sweep 7 w-tiles of 16 pixels, reusing A + bias
    for (int wt = 0; wt < 7; ++wt) {
        const int col = wt * 16 + m;                    // output column for this lane

        // B fragment: im2col rows k = 4c+halfk, 4c+halfk+1 for pixel `col`
        v2f bm[3];
#pragma unroll
        for (int c = 0; c < 3; ++c) {
            const int k0 = c * 4 + halfk;
            const int kx = (k0     < 9) ? k0     : 8;
            const int ky = (k0 + 1 < 9) ? k0 + 1 : 8;
            const int khx = kx / 3, kwx = kx - khx * 3;
            const int khy = ky / 3, kwy = ky - khy * 3;
            float bx = xplane[(size_t)(h + khx) * HPAD + col + kwx];
            float by = xplane[(size_t)(h + khy) * HPAD + col + kwy];
            bm[c].x = (k0     < 9) ? bx : 0.0f;
            bm[c].y = (k0 + 1 < 9) ? by : 0.0f;
        }

        v8f d = cbias;
        d = __builtin_amdgcn_wmma_f32_16x16x4_f32(false, a[0], false, bm[0],
                                                  (short)0, d, false, false);
        d = __builtin_amdgcn_wmma_f32_16x16x4_f32(false, a[1], false, bm[1],
                                                  (short)0, d, false, false);
        d = __builtin_amdgcn_wmma_f32_16x16x4_f32(false, a[2], false, bm[2],
                                                  (short)0, d, false, false);

        // ---- Store: D element r -> out[b][coHalf+r][h][col].
        // 205 MB of streaming writes dominate the roofline -> non-temporal so the
        // output doesn't evict the L2-resident input plane.
#pragma unroll
        for (int r = 0; r < 8; ++r) {
            float* p = out + (((size_t)(b * COUT + coHalf + r) * HOUT) + h) * HOUT + col;
            __builtin_nontemporal_store(d[r], p);
        }
    }
}

extern "C" void kernel_launch(void* const* d_in, const int* in_sizes, int n_in,
                              void* d_out, int out_size, void* d_ws, size_t ws_size,
                              hipStream_t stream) {
    const float* x    = (const float*)d_in[0];   // x_padded (64,64,114,114) f32
    const float* wgt  = (const float*)d_in[1];   // weight   (64,64,3,3)     f32
    const float* bias = (const float*)d_in[2];   // bias     (64,)           f32
    float*       out  = (float*)d_out;           // (64,64,112,112)          f32

    // waves = 64 (b) * 4 (co tiles) * 112 (rows) = 28672; 8 waves/block
    const int blocks = (64 * 4 * HOUT) / 8;      // 3584
    conv3x3_lastchan_wmma<<<dim3(blocks), dim3(256), 0, stream>>>(x, wgt, bias, out);
}